// ScaledDotProductAttention_39109972198042
// MI455X (gfx1250) — compile-verified
//
#include <hip/hip_runtime.h>
#include <hip/hip_bf16.h>
#include <stdint.h>

// Problem constants (match reference)
#define BB 2
#define HH 16
#define TT 2048
#define SS 2048
#define DD 64
#define TTILE 16              // query rows per workgroup
#define NWAVES 4              // 128 threads, wave32
#define SQ_STRIDE 65          // padded Q tile row stride (floats)
#define SS_STRIDE 2049        // padded score row stride (floats) -> conflict-free lane%16 gathers

typedef __attribute__((ext_vector_type(2))) float v2f;
typedef __attribute__((ext_vector_type(8))) float v8f;

// LDS layout (floats)
#define SQ_OFF   0
#define SS_OFF   (TTILE * SQ_STRIDE)
#define RED_OFF  (SS_OFF + TTILE * SS_STRIDE)
#define MAX_OFF  (RED_OFF + TTILE * 8)
#define SUM_OFF  (MAX_OFF + TTILE)
#define LDS_FLOATS (SUM_OFF + TTILE)

__global__ __launch_bounds__(NWAVES * 32, 1)
void sdpa_fp32_wmma_kernel(const float* __restrict__ q,
                           const float* __restrict__ k,
                           const float* __restrict__ v,
                           const unsigned char* __restrict__ mask,
                           float* __restrict__ ctx_out,
                           float* __restrict__ attn_out) {
    extern __shared__ float smem[];
    float* sQ   = smem + SQ_OFF;
    float* sS   = smem + SS_OFF;
    float* sRed = smem + RED_OFF;
    float* sMax = smem + MAX_OFF;
    float* sSum = smem + SUM_OFF;

    const int tid  = threadIdx.x;
    const int wave = tid >> 5;
    const int lane = tid & 31;

    // block -> (bh, query tile)
    const int blk    = blockIdx.x;              // 0 .. B*H*(T/16)-1
    const int tTiles = TT / TTILE;              // 128
    const int bh     = blk / tTiles;            // 0 .. 31
    const int tBase  = (blk % tTiles) * TTILE;
    const int bIdx   = bh / HH;                 // batch (mask has no H dim)

    const float scale = 0.125f;                 // 1/sqrt(64)

    const float* qp = q + (size_t)bh * TT * DD;
    const float* kp = k + (size_t)bh * SS * DD;
    const float* vp = v + (size_t)bh * SS * DD;
    const unsigned char* mp = mask + (size_t)bIdx * TT * SS;

    // ---- Phase 1: stage Q tile (scaled) into LDS, coalesced ----
    for (int e = tid; e < TTILE * DD; e += NWAVES * 32) {
        int r = e >> 6;          // /DD
        int d = e & (DD - 1);
        sQ[r * SQ_STRIDE + d] = qp[(size_t)(tBase + r) * DD + d] * scale;
    }
    __syncthreads();

    // Per-lane WMMA fragment coordinates (32-bit A/B 16x4 layout)
    const int mn   = lane & 15;                 // A: row m, B: col n
    const int koff = (lane >> 4) << 1;          // 0 or 2
    const int mOff = (lane >> 4) << 3;          // C: 0 or 8

    // ---- Phase 2: scores S = (Q*scale) @ K^T, two s-tiles in flight per wave ----
    const int sTiles = SS / 16;                 // 128
    for (int t2 = wave * 2; t2 < sTiles; t2 += NWAVES * 2) {
        const int sBase0 = t2 * 16;
        const int sBase1 = sBase0 + 16;
        v8f c0 = {}, c1 = {};
        const float* kRow0 = kp + (size_t)(sBase0 + mn) * DD;
        const float* kRow1 = kp + (size_t)(sBase1 + mn) * DD;
        for (int kk = 0; kk < DD; kk += 4) {
            v2f a, b0, b1;
            a.x  = sQ[mn * SQ_STRIDE + kk + koff];
            a.y  = sQ[mn * SQ_STRIDE + kk + koff + 1];
            b0.x = kRow0[kk + koff];
            b0.y = kRow0[kk + koff + 1];
            b1.x = kRow1[kk + koff];
            b1.y = kRow1[kk + koff + 1];
            c0 = __builtin_amdgcn_wmma_f32_16x16x4_f32(
                    false, a, false, b0, (short)0, c0, false, false);
            c1 = __builtin_amdgcn_wmma_f32_16x16x4_f32(
                    false, a, false, b1, (short)0, c1, false, false);
        }
        for (int j = 0; j < 8; ++j) {
            const int m = j + mOff;
            sS[m * SS_STRIDE + sBase0 + mn] = c0[j];
            sS[m * SS_STRIDE + sBase1 + mn] = c1[j];
        }
    }
    __syncthreads();

    // ---- Phase 2b: coalesced mask application (vectorized 4 bytes/lane) ----
    {
        const uint32_t* m32 = (const uint32_t*)(mp + (size_t)tBase * SS);
        for (int e4 = tid; e4 < (TTILE * SS) / 4; e4 += NWAVES * 32) {
            const uint32_t mk = m32[e4];
            if (mk != 0x01010101u) {            // fast path: all-true word
                const int e = e4 * 4;
                for (int bix = 0; bix < 4; ++bix) {
                    if (((mk >> (8 * bix)) & 0xFFu) == 0u) {
                        const int ee = e + bix;
                        sS[(ee >> 11) * SS_STRIDE + (ee & (SS - 1))] = -__builtin_inff();
                    }
                }
            }
        }
    }
    __syncthreads();

    // ---- Phase 3: row softmax over S=2048 (8 threads per row) ----
    const int row = tid >> 3;                   // 0..15
    const int sub = tid & 7;                    // 0..7
    {
        float pm = -__builtin_inff();
        for (int cix = sub; cix < SS; cix += 8) {
            pm = fmaxf(pm, sS[row * SS_STRIDE + cix]);
        }
        sRed[row * 8 + sub] = pm;
    }
    __syncthreads();
    if (tid < TTILE) {
        float m0 = sRed[tid * 8];
        for (int i = 1; i < 8; ++i) m0 = fmaxf(m0, sRed[tid * 8 + i]);
        sMax[tid] = m0;
    }
    __syncthreads();
    {
        const float rm = sMax[row];
        float ps = 0.0f;
        for (int cix = sub; cix < SS; cix += 8) {
            float e = __expf(sS[row * SS_STRIDE + cix] - rm);
            sS[row * SS_STRIDE + cix] = e;
            ps += e;
        }
        sRed[row * 8 + sub] = ps;
    }
    __syncthreads();
    if (tid < TTILE) {
        float s0 = 0.0f;
        for (int i = 0; i < 8; ++i) s0 += sRed[tid * 8 + i];
        sSum[tid] = s0;
    }
    __syncthreads();

    // ---- Phase 3b: normalize in LDS + coalesced attn writeback (512MB stream) ----
    {
        float* aOut = attn_out + ((size_t)bh * TT + tBase) * SS;
        for (int e = tid; e < TTILE * SS; e += NWAVES * 32) {
            const int r   = e >> 11;            // /SS
            const int cix = e & (SS - 1);
            const float p = sS[r * SS_STRIDE + cix] / sSum[r];
            sS[r * SS_STRIDE + cix] = p;
            aOut[e] = p;
        }
    }
    __syncthreads();

    // ---- Phase 4: ctx = P @ V; wave w owns D cols [16w,16w+16); K-split dual acc ----
    {
        const int dBase = wave * 16;
        v8f c0 = {}, c1 = {};
        for (int sk = 0; sk < SS; sk += 8) {
            v2f a0, b0, a1, b1;
            a0.x = sS[mn * SS_STRIDE + sk + koff];
            a0.y = sS[mn * SS_STRIDE + sk + koff + 1];
            b0.x = vp[(size_t)(sk + koff) * DD + dBase + mn];
            b0.y = vp[(size_t)(sk + koff + 1) * DD + dBase + mn];
            a1.x = sS[mn * SS_STRIDE + sk + 4 + koff];
            a1.y = sS[mn * SS_STRIDE + sk + 4 + koff + 1];
            b1.x = vp[(size_t)(sk + 4 + koff) * DD + dBase + mn];
            b1.y = vp[(size_t)(sk + 4 + koff + 1) * DD + dBase + mn];
            c0 = __builtin_amdgcn_wmma_f32_16x16x4_f32(
                    false, a0, false, b0, (short)0, c0, false, false);
            c1 = __builtin_amdgcn_wmma_f32_16x16x4_f32(
                    false, a1, false, b1, (short)0, c1, false, false);
        }
        for (int j = 0; j < 8; ++j) {
            const int m = j + mOff;
            ctx_out[(size_t)(bh * TT + tBase + m) * DD + dBase + mn] = c0[j] + c1[j];
        }
    }
}

extern "C" void kernel_launch(void* const* d_in, const int* in_sizes, int n_in,
                              void* d_out, int out_size, void* d_ws, size_t ws_size,
                              hipStream_t stream) {
    (void)in_sizes; (void)n_in; (void)d_ws; (void)ws_size; (void)out_size;
    const float* q = (const float*)d_in[0];
    const float* k = (const float*)d_in[1];
    const float* v = (const float*)d_in[2];
    const unsigned char* mask = (const unsigned char*)d_in[3];

    float* ctx_out  = (float*)d_out;                                    // [B,H,T,D]
    float* attn_out = ctx_out + (size_t)BB * HH * TT * DD;              // [B,H,T,S]

    const int blocks = BB * HH * (TT / TTILE);                          // 4096
    const size_t lds_bytes = (size_t)LDS_FLOATS * sizeof(float);        // ~133 KB

    sdpa_fp32_wmma_kernel<<<blocks, NWAVES * 32, lds_bytes, stream>>>(
        q, k, v, mask, ctx_out, attn_out);
}